// WindowSelfAttention_41266045780056
// MI455X (gfx1250) — compile-verified
//
#include <hip/hip_runtime.h>
#include <math.h>

#define DIM 384
#define QKV3 (3*DIM)
#define HEADS 12
#define HD 32
#define NTOK 49
#define NP 64
#define XS_STRIDE 388   // %64==4: conflict-free strided reads; %4==0: float4 stores
#define WT_STRIDE 388   // transposed W, same conflict-free geometry
#define QV_STRIDE 36    // gcd(36,64)=4 -> 16-row cycle conflict-free, b64-aligned
#define VT_STRIDE 68
#define SM_STRIDE 68    // %64==4 -> conflict-free

typedef __attribute__((ext_vector_type(2))) float v2f;
typedef __attribute__((ext_vector_type(8))) float v8f;

// Pre-transposed weights (filled once per launch by transpose_kernel).
__device__ float g_wqt[QKV3 * DIM];   // w_qkv^T : g_wqt[col][k], col in 0..1151
__device__ float g_wpt[DIM * DIM];    // w_proj^T: g_wpt[n][k]

// D = A(16x4) * B(4x16) + C, full fp32 (matches reference precision)
__device__ __forceinline__ v8f wmma4(v2f a, v2f b, v8f c) {
  return __builtin_amdgcn_wmma_f32_16x16x4_f32(false, a, false, b, (short)0, c,
                                               false, false);
}

// ---------------------------------------------------------------------------
// Kernel 0: one-time weight transposes (coalesced writes).
// ---------------------------------------------------------------------------
__global__ __launch_bounds__(256) void transpose_kernel(
    const float* __restrict__ w_qkv, const float* __restrict__ w_proj)
{
  int idx = blockIdx.x * 256 + threadIdx.x;
  if (idx < QKV3 * DIM) {
    int col = idx / DIM, k = idx - col * DIM;
    g_wqt[idx] = w_qkv[k * QKV3 + col];
  }
  if (idx < DIM * DIM) {
    int n = idx / DIM, k = idx - n * DIM;
    g_wpt[idx] = w_proj[k * DIM + n];
  }
}

// ---------------------------------------------------------------------------
// Kernel 1: per (window, head) fused  qkv-projection -> q.k^T+bias -> softmax
//           -> P.v, result written to workspace y[win][tok][head*32+d]
// 256 threads = 8 waves. LDS ~293KB (1 block/WGP, 2 waves/SIMD).
// ---------------------------------------------------------------------------
__global__ __launch_bounds__(256) void swin_attn_kernel(
    const float* __restrict__ x, const float* __restrict__ b_qkv,
    const float* __restrict__ table, const int* __restrict__ relidx,
    float* __restrict__ y)
{
  extern __shared__ float lds[];
  float* xs = lds;                          // 64 x 388   (x window, rows padded)
  float* wt = xs + NP * XS_STRIDE;          // 96 x 388   (W^T: wt[col][k])
  float* qs = wt + 96 * WT_STRIDE;          // 64 x 36
  float* ks = qs + NP * QV_STRIDE;          // 64 x 36
  float* vt = ks + NP * QV_STRIDE;          // 32 x 68    (v^T: vt[d][token])
  float* sm = vt + HD * VT_STRIDE;          // 64 x 68

  const int head = blockIdx.x;
  const int win  = blockIdx.y;
  const int tid  = threadIdx.x;
  const int wave = tid >> 5;                // 0..7
  const int lane = tid & 31;
  const int hf   = lane >> 4;               // lane half: K pair select / M+8 rows
  const int l16  = lane & 15;

  const float* xw = x + (size_t)win * (NTOK * DIM);

  // ---- stage x window (float4), zero-pad rows 49..63 ----
  for (int i4 = tid; i4 < (NTOK * DIM) / 4; i4 += 256) {
    int r  = i4 / (DIM / 4);
    int c4 = (i4 - r * (DIM / 4)) * 4;
    *(float4*)(xs + r * XS_STRIDE + c4) = *(const float4*)(xw + r * DIM + c4);
  }
  for (int i4 = tid; i4 < ((NP - NTOK) * DIM) / 4; i4 += 256) {
    int r  = NTOK + i4 / (DIM / 4);
    int c4 = (i4 % (DIM / 4)) * 4;
    *(float4*)(xs + r * XS_STRIDE + c4) = make_float4(0.f, 0.f, 0.f, 0.f);
  }
  // ---- stage this head's W^T slice from g_wqt: b128 -> b128, conflict-free
  for (int i4 = tid; i4 < 96 * (DIM / 4); i4 += 256) {
    int colL = i4 / (DIM / 4);
    int k4   = (i4 - colL * (DIM / 4)) * 4;
    int s = colL >> 5, c = colL & 31;
    *(float4*)(wt + colL * WT_STRIDE + k4) =
        *(const float4*)(&g_wqt[(s * DIM + head * HD + c) * DIM + k4]);
  }
  __syncthreads();

  // ---- Step A: q/k/v = x @ W + b. 24 tiles; wave owns M-band, 3 (s,nt) combos
  {
    const int mt = wave & 3;
    const int grpbase = (wave >> 2) * 3;
    const int m0 = mt * 16;
    const float* ap = xs + (m0 + l16) * XS_STRIDE + 2 * hf;
    const float* bp[3]; float bias[3]; v8f acc[3];
    const v8f vzero = {};
    #pragma unroll
    for (int j = 0; j < 3; ++j) {
      int jj = grpbase + j;
      int s = jj >> 1, nt = jj & 1;
      bp[j]   = wt + (s * 32 + nt * 16 + l16) * WT_STRIDE + 2 * hf;
      bias[j] = b_qkv[s * DIM + head * HD + nt * 16 + l16];
      acc[j]  = vzero;
    }
    #pragma unroll 4
    for (int k0 = 0; k0 < DIM; k0 += 4) {
      v2f a = *(const v2f*)(ap + k0);       // one ds_load_b64 feeds 3 WMMAs
      #pragma unroll
      for (int j = 0; j < 3; ++j) {
        v2f b = *(const v2f*)(bp[j] + k0);  // ds_load_b64 / 2addr_b64 merged
        acc[j] = wmma4(a, b, acc[j]);
      }
    }
    #pragma unroll
    for (int j = 0; j < 3; ++j) {
      int jj = grpbase + j;
      int s = jj >> 1, nt = jj & 1;
      if (s == 2) {                         // v stored transposed: vt[d][token]
        float* vd = vt + (nt * 16 + l16) * VT_STRIDE + m0 + 8 * hf;
        #pragma unroll
        for (int r = 0; r < 8; ++r) vd[r] = acc[j][r] + bias[j];
      } else {
        float* dqk = (s == 0 ? qs : ks) + nt * 16 + l16;
        #pragma unroll
        for (int r = 0; r < 8; ++r)
          dqk[(m0 + r + 8 * hf) * QV_STRIDE] = acc[j][r] + bias[j];
      }
    }
  }
  __syncthreads();

  // ---- Step B: S = q @ k^T * scale + bias. 16 tiles; wave = (mt, 2 nt) ----
  const float scale = 0.17677669529663687f;   // 32^-0.5
  {
    const int mt = wave & 3;
    const int ng = wave >> 2;
    const int m0 = mt * 16;
    const float* qp  = qs + (m0 + l16) * QV_STRIDE + 2 * hf;
    const float* kp0 = ks + (ng * 32 + l16) * QV_STRIDE + 2 * hf;
    const float* kp1 = ks + (ng * 32 + 16 + l16) * QV_STRIDE + 2 * hf;
    v8f a0v = {}, a1v = {};
    #pragma unroll
    for (int k0 = 0; k0 < HD; k0 += 4) {
      v2f a  = *(const v2f*)(qp  + k0);
      v2f b0 = *(const v2f*)(kp0 + k0);
      v2f b1 = *(const v2f*)(kp1 + k0);
      a0v = wmma4(a, b0, a0v);
      a1v = wmma4(a, b1, a1v);
    }
    v8f accs[2] = { a0v, a1v };
    #pragma unroll
    for (int j = 0; j < 2; ++j) {
      int n0 = (2 * ng + j) * 16;
      #pragma unroll
      for (int r = 0; r < 8; ++r) {
        int row = m0 + r + 8 * hf;
        int cv  = n0 + l16;
        float val;
        if (row < NTOK && cv < NTOK)
          val = accs[j][r] * scale + table[relidx[row * NTOK + cv] * HEADS + head];
        else
          val = -1e30f;                      // mask padded columns
        sm[row * SM_STRIDE + cv] = val;
      }
    }
  }
  __syncthreads();

  // ---- softmax over rows 0..48 ----
  if (tid < NTOK) {
    float* rowp = sm + tid * SM_STRIDE;
    float mx = -1e30f;
    for (int c = 0; c < NTOK; ++c) mx = fmaxf(mx, rowp[c]);
    float sum = 0.f;
    for (int c = 0; c < NTOK; ++c) sum += __expf(rowp[c] - mx);
    float inv = 1.f / sum;
    for (int c = 0; c < NTOK; ++c) rowp[c] = __expf(rowp[c] - mx) * inv;
    for (int c = NTOK; c < NP; ++c) rowp[c] = 0.f;  // padded K contributes 0
  }
  __syncthreads();

  // ---- Step C: O = P @ v. 8 tiles; one per wave, split-K dual accum ----
  {
    const int mt = wave & 3;
    const int nt = wave >> 2;
    const int m0 = mt * 16, n0 = nt * 16;
    const float* pp = sm + (m0 + l16) * SM_STRIDE + 2 * hf;
    const float* vp = vt + (n0 + l16) * VT_STRIDE + 2 * hf;   // v^T: contiguous K
    v8f c0 = {}, c1 = {};
    #pragma unroll
    for (int k0 = 0; k0 < NP; k0 += 8) {
      v2f a0 = *(const v2f*)(pp + k0);
      v2f b0 = *(const v2f*)(vp + k0);
      v2f a1 = *(const v2f*)(pp + k0 + 4);
      v2f b1 = *(const v2f*)(vp + k0 + 4);
      c0 = wmma4(a0, b0, c0);
      c1 = wmma4(a1, b1, c1);
    }
    v8f c = c0 + c1;
    float* yw = y + (size_t)win * (NTOK * DIM) + head * HD;
    #pragma unroll
    for (int r = 0; r < 8; ++r) {
      int row = m0 + r + 8 * hf;
      if (row < NTOK)
        yw[row * DIM + n0 + l16] = c[r];
    }
  }
}

// ---------------------------------------------------------------------------
// Kernel 2: out = y @ w_proj + b_proj over flattened [M,384]x[384,384].
// 256 threads = 8 waves; wave owns one M-band, 12 N-tiles (2 groups of 6).
// B fragments are contiguous global b64 from pre-transposed g_wpt (L2-hot).
// ---------------------------------------------------------------------------
__global__ __launch_bounds__(256) void swin_proj_kernel(
    const float* __restrict__ y, const float* __restrict__ b_proj,
    float* __restrict__ out, int M)
{
  extern __shared__ float lds[];
  float* as = lds;                          // 64 x 388
  const int rowbase = blockIdx.x * 64;
  const int tid  = threadIdx.x;
  const int wave = tid >> 5;
  const int lane = tid & 31;
  const int hf   = lane >> 4;
  const int l16  = lane & 15;
  const int nrows = min(64, M - rowbase);

  for (int i4 = tid; i4 < (64 * DIM) / 4; i4 += 256) {
    int r  = i4 / (DIM / 4);
    int c4 = (i4 % (DIM / 4)) * 4;
    float4 v = make_float4(0.f, 0.f, 0.f, 0.f);
    if (r < nrows) v = *(const float4*)(y + (size_t)(rowbase + r) * DIM + c4);
    *(float4*)(as + r * XS_STRIDE + c4) = v;
  }
  __syncthreads();

  const int mt = wave & 3;
  const int nh = wave >> 2;
  const int m0 = mt * 16;
  const float* ap = as + (m0 + l16) * XS_STRIDE + 2 * hf;

  for (int g = 0; g < 2; ++g) {
    const float* bp[6]; float bias[6]; v8f acc[6];
    const v8f vzero = {};
    #pragma unroll
    for (int j = 0; j < 6; ++j) {
      int n0 = (nh * 12 + g * 6 + j) * 16;
      bp[j]   = &g_wpt[(n0 + l16) * DIM + 2 * hf];
      bias[j] = b_proj[n0 + l16];
      acc[j]  = vzero;
    }
    #pragma unroll 2
    for (int k0 = 0; k0 < DIM; k0 += 4) {
      v2f a = *(const v2f*)(ap + k0);       // one ds_load_b64 feeds 6 WMMAs
      #pragma unroll
      for (int j = 0; j < 6; ++j) {
        v2f b = *(const v2f*)(bp[j] + k0);  // single global_load_b64, imm offset
        acc[j] = wmma4(a, b, acc[j]);
      }
    }
    #pragma unroll
    for (int j = 0; j < 6; ++j) {
      int n0 = (nh * 12 + g * 6 + j) * 16;
      #pragma unroll
      for (int r = 0; r < 8; ++r) {
        int row = m0 + r + 8 * hf;
        if (row < nrows)
          out[(size_t)(rowbase + row) * DIM + n0 + l16] = acc[j][r] + bias[j];
      }
    }
  }
}

extern "C" void kernel_launch(void* const* d_in, const int* in_sizes, int n_in,
                              void* d_out, int out_size, void* d_ws, size_t ws_size,
                              hipStream_t stream) {
  const float* x      = (const float*)d_in[0];
  const float* w_qkv  = (const float*)d_in[1];
  const float* b_qkv  = (const float*)d_in[2];
  const float* w_proj = (const float*)d_in[3];
  const float* b_proj = (const float*)d_in[4];
  const float* table  = (const float*)d_in[5];
  const int*   relidx = (const int*)d_in[6];

  const int num_win = in_sizes[0] / (NTOK * DIM);
  float* y = (float*)d_ws;                  // [num_win * 49 * 384] fp32 staging

  const size_t lds1 =
      (size_t)(NP * XS_STRIDE + 96 * WT_STRIDE + 2 * NP * QV_STRIDE +
               HD * VT_STRIDE + NP * SM_STRIDE) * sizeof(float);
  const size_t lds2 = (size_t)(64 * XS_STRIDE) * sizeof(float);

  (void)hipFuncSetAttribute((const void*)swin_attn_kernel,
                            hipFuncAttributeMaxDynamicSharedMemorySize, (int)lds1);
  (void)hipFuncSetAttribute((const void*)swin_proj_kernel,
                            hipFuncAttributeMaxDynamicSharedMemorySize, (int)lds2);

  // one-time weight transposes into static device buffers
  transpose_kernel<<<(QKV3 * DIM + 255) / 256, 256, 0, stream>>>(w_qkv, w_proj);

  dim3 g1(HEADS, num_win);
  swin_attn_kernel<<<g1, 256, lds1, stream>>>(x, b_qkv, table, relidx, y);

  const int M = num_win * NTOK;
  swin_proj_kernel<<<(M + 63) / 64, 256, lds2, stream>>>(y, b_proj,
                                                         (float*)d_out, M);
}